// UltimusBlock_13984413516170
// MI455X (gfx1250) — compile-verified
//
#include <hip/hip_runtime.h>
#include <math.h>

// ---------------------------------------------------------------------------
// MI455X (gfx1250) "Ultimus" block.
//   Pass1: KQV via V_WMMA_F32_16X16X4_F32, V -> workspace, Q^T K partials
//          (x staged to LDS with global_load_async_to_lds_b128, double buffered)
//   Mid  : reduce partials, softmax/sqrt(8), M2 = AM @ Wo^T
//   Pass2: OUT = V @ M2 + bo via WMMA
// Roofline: ~448 MB traffic (x 192R + V 32W + V 32R + out 192W) -> ~19 us
// @ 23.3 TB/s. x/out get NT hints so the 32 MB V set stays in the 192 MB L2.
// ---------------------------------------------------------------------------

typedef float v2f __attribute__((ext_vector_type(2)));
typedef float v4f __attribute__((ext_vector_type(4)));
typedef float v8f __attribute__((ext_vector_type(8)));
typedef int   v4i __attribute__((vector_size(16)));   // matches builtin proto

#define N_ROWS    1048576
#define N_TILES   (N_ROWS / 16)      // 65536 tiles of 16 rows
#define BLOCKS_MAIN 1024
#define WPB         8                // waves per block (256 threads, wave32)
#define TOTAL_WAVES (BLOCKS_MAIN * WPB)

#define LDS_AS __attribute__((address_space(3)))

#if __has_builtin(__builtin_amdgcn_global_load_async_to_lds_b128)
#define HAVE_ASYNC_LDS 1
#define ASYNC_B128(gsrc, ldst, cpol)                                      \
  __builtin_amdgcn_global_load_async_to_lds_b128(                         \
      (v4i*)(gsrc), (LDS_AS v4i*)(ldst), 0, (cpol))
#else
#define HAVE_ASYNC_LDS 0
#endif

#if __has_builtin(__builtin_amdgcn_s_wait_asynccnt)
#define WAIT_ASYNC(n) __builtin_amdgcn_s_wait_asynccnt(n)
#else
#define WAIT_ASYNC(n) asm volatile("s_wait_asynccnt %0" ::"i"(n))
#endif

__device__ __forceinline__ v8f wmma_f32(v2f a, v2f b, v8f c) {
  // D = A(16x4) * B(4x16) + C(16x16), fp32 throughout.
  return __builtin_amdgcn_wmma_f32_16x16x4_f32(
      false, a, false, b, (short)0, c, false, false);
}

__device__ __forceinline__ float shfl32(float v, int src) {
  return __shfl(v, src & 31, 32);
}

// ---------------------------------------------------------------------------
// Pass 1
// ---------------------------------------------------------------------------

// Stage one 16x48 f32 tile of x into LDS (rows padded to 50 dwords).
__device__ __forceinline__ void stage_x(const float* xt, float* dst, int lane) {
#pragma unroll
  for (int i = 0; i < 6; ++i) {
    const int idx  = i * 32 + lane;          // float4 index 0..191
    const int flat = idx * 4;                // 48 % 4 == 0 -> stays in-row
    const int r = flat / 48, c = flat % 48;
#if HAVE_ASYNC_LDS
    // async DMA, NT hint (cpol TH=NT): x is a pure 192MB stream
    ASYNC_B128(xt + flat, dst + r * 50 + c, 1);
#else
    const v4f v = __builtin_nontemporal_load((const v4f*)xt + idx);
    float* d = &dst[r * 50 + c];
    d[0] = v.x; d[1] = v.y; d[2] = v.z; d[3] = v.w;
#endif
  }
}

__global__ __launch_bounds__(256) void pass1(
    const float* __restrict__ x,
    const float* __restrict__ Wk, const float* __restrict__ bk,
    const float* __restrict__ Wq, const float* __restrict__ bq,
    const float* __restrict__ Wv, const float* __restrict__ bv,
    float* __restrict__ Vws, float* __restrict__ partials)
{
  const int lane = threadIdx.x & 31;
  const int wave = threadIdx.x >> 5;
  const int wgid = blockIdx.x * WPB + wave;
  const int hi   = lane >> 4;        // lane half (0/1)
  const int m    = lane & 15;        // row (A) / col (B,C) index

  // ---- B fragments for fused [K|Q] and [V|0] weights --------------------
  // B 4x16 layout: v.x = row (4kc+2*hi), v.y = row (4kc+2*hi+1), col = m.
  v2f bKQ[12], bV[12];
#pragma unroll
  for (int kc = 0; kc < 12; ++kc) {
    const int c0 = 4 * kc + 2 * hi;
    float kq0, kq1, vv0 = 0.f, vv1 = 0.f;
    if (m < 8) {
      kq0 = Wk[m * 48 + c0];       kq1 = Wk[m * 48 + c0 + 1];
      vv0 = Wv[m * 48 + c0];       vv1 = Wv[m * 48 + c0 + 1];
    } else {
      kq0 = Wq[(m - 8) * 48 + c0]; kq1 = Wq[(m - 8) * 48 + c0 + 1];
    }
    bKQ[kc].x = kq0; bKQ[kc].y = kq1;
    bV[kc].x  = vv0; bV[kc].y  = vv1;
  }
  const float bias0 = (m < 8) ? bk[m] : bq[m - 8];
  const float bias1 = (m < 8) ? bv[m] : 0.f;

  // ---- double-buffered per-wave LDS tile (16 rows x 50 dwords) ----------
  __shared__ float xs[WPB][2 * 800];
  float* buf0 = &xs[wave][0];
  float* buf1 = &xs[wave][800];

  v8f sacc = {0.f, 0.f, 0.f, 0.f, 0.f, 0.f, 0.f, 0.f};   // Q^T K accumulator

  if (wgid < N_TILES) stage_x(x + (size_t)wgid * 768, buf0, lane);

  int parity = 0;
  for (int tile = wgid; tile < N_TILES; tile += TOTAL_WAVES) {
    float* cur = parity ? buf1 : buf0;
    float* nxt = parity ? buf0 : buf1;
    const int nt = tile + TOTAL_WAVES;
    if (nt < N_TILES) {
      stage_x(x + (size_t)nt * 768, nxt, lane);   // overlap next tile
#if HAVE_ASYNC_LDS
      WAIT_ASYNC(6);                              // current tile's 6 ops done
#endif
    } else {
#if HAVE_ASYNC_LDS
      WAIT_ASYNC(0);
#endif
    }
    parity ^= 1;
#if !HAVE_ASYNC_LDS
    const int pf = tile + 2 * TOTAL_WAVES;
    if (pf < N_TILES)
      __builtin_prefetch(x + (size_t)pf * 768 + lane * 24, 0, 3);
#endif

    v8f acc0, acc1;                               // [K|Q], [V|0] (+bias)
#pragma unroll
    for (int i = 0; i < 8; ++i) { acc0[i] = bias0; acc1[i] = bias1; }

    // ---- KQV GEMM: 12 K-chunks, A from LDS (16x4 layout) ----------------
#pragma unroll
    for (int kc = 0; kc < 12; ++kc) {
      v2f a = *(const v2f*)&cur[m * 50 + 4 * kc + 2 * hi];
      acc0 = wmma_f32(a, bKQ[kc], acc0);
      acc1 = wmma_f32(a, bV[kc], acc1);
    }

    // ---- store V tile row-major (kept RT so it stays L2-resident) -------
    if (m < 8) {
      float* vb = Vws + (size_t)tile * 128;
#pragma unroll
      for (int v = 0; v < 8; ++v) vb[(v + 8 * hi) * 8 + m] = acc1[v];
    }

    // ---- scores += Q^T K : C-layout -> A/B frags via shuffles -----------
    // Q[r,h] at lane 8+h+16*(r>>3), reg r&7 ; K[r,k] at lane k+16*(r>>3)
#pragma unroll
    for (int c = 0; c < 4; ++c) {
      const int r0 = 4 * c, r1 = 4 * c + 1, r2 = 4 * c + 2, r3 = 4 * c + 3;
      float qa = shfl32(acc0[r0 & 7], 8 + m + ((r0 >> 3) << 4));
      float qb = shfl32(acc0[r2 & 7], 8 + m + ((r2 >> 3) << 4));
      float qc = shfl32(acc0[r1 & 7], 8 + m + ((r1 >> 3) << 4));
      float qd = shfl32(acc0[r3 & 7], 8 + m + ((r3 >> 3) << 4));
      v2f aS; aS.x = hi ? qb : qa; aS.y = hi ? qd : qc;
      float ka  = shfl32(acc0[r0 & 7], m + ((r0 >> 3) << 4));
      float kb  = shfl32(acc0[r2 & 7], m + ((r2 >> 3) << 4));
      float kc2 = shfl32(acc0[r1 & 7], m + ((r1 >> 3) << 4));
      float kd  = shfl32(acc0[r3 & 7], m + ((r3 >> 3) << 4));
      v2f bS; bS.x = hi ? kb : ka; bS.y = hi ? kd : kc2;
      sacc = wmma_f32(aS, bS, sacc);              // valid region rows/cols 0..7
    }
  }

  // ---- block-level score reduction: flat index k*8+h --------------------
  __shared__ float sred[WPB][64];
  if (lane < 8) {
#pragma unroll
    for (int v = 0; v < 8; ++v) sred[wave][lane * 8 + v] = sacc[v];
  }
  __syncthreads();
  if (threadIdx.x < 64) {
    float s = 0.f;
#pragma unroll
    for (int w = 0; w < WPB; ++w) s += sred[w][threadIdx.x];
    partials[blockIdx.x * 64 + threadIdx.x] = s;
  }
}

// ---------------------------------------------------------------------------
// Mid: reduce partials (8-way sliced), softmax(axis=k)/sqrt(8), M2 = AM @ Wo^T
// ---------------------------------------------------------------------------
__global__ void pass_mid(const float* __restrict__ partials,
                         const float* __restrict__ Wo,
                         float* __restrict__ M2)
{
  __shared__ float psum[8][64];
  __shared__ float sc[64];   // flat[k*8 + h]
  __shared__ float AM[64];   // [h*8 + k]
  const int t = threadIdx.x;               // 512 threads
  const int e = t & 63, s = t >> 6;        // entry, slice
  {
    float v = 0.f;
    for (int p = s; p < BLOCKS_MAIN; p += 8) v += partials[p * 64 + e];
    psum[s][e] = v;
  }
  __syncthreads();
  if (t < 64) {
    float v = 0.f;
#pragma unroll
    for (int w = 0; w < 8; ++w) v += psum[w][t];
    sc[t] = v;
  }
  __syncthreads();
  if (t < 8) {               // h = t, softmax over k
    float mx = -3.4e38f;
#pragma unroll
    for (int k = 0; k < 8; ++k) mx = fmaxf(mx, sc[k * 8 + t]);
    float ex[8], den = 0.f;
#pragma unroll
    for (int k = 0; k < 8; ++k) { ex[k] = __expf(sc[k * 8 + t] - mx); den += ex[k]; }
    const float inv = 1.f / (den * sqrtf(8.f));
#pragma unroll
    for (int k = 0; k < 8; ++k) AM[t * 8 + k] = ex[k] * inv;
  }
  __syncthreads();
  if (t < 384) {             // M2[h,c] = sum_k AM[h,k] * Wo[c,k]
    const int c = t / 8, h = t % 8;
    float v = 0.f;
#pragma unroll
    for (int k = 0; k < 8; ++k) v += AM[h * 8 + k] * Wo[c * 8 + k];
    M2[h * 48 + c] = v;
  }
}

// ---------------------------------------------------------------------------
// Pass 2: OUT = V @ M2 + bo   ([N,8] x [8,48]), 6 WMMAs / 16-row tile
// ---------------------------------------------------------------------------
__global__ __launch_bounds__(256) void pass2(
    const float* __restrict__ Vws, const float* __restrict__ M2,
    const float* __restrict__ bo, float* __restrict__ out)
{
  const int lane = threadIdx.x & 31;
  const int wave = threadIdx.x >> 5;
  const int wgid = blockIdx.x * WPB + wave;
  const int hi   = lane >> 4;
  const int m    = lane & 15;

  // B fragments: 3 column tiles x 2 K-chunks of M2 [8 x 48]
  v2f bO[3][2];
  float biasv[3];
#pragma unroll
  for (int ct = 0; ct < 3; ++ct) {
#pragma unroll
    for (int c = 0; c < 2; ++c) {
      const int r = 4 * c + 2 * hi;
      bO[ct][c].x = M2[r * 48 + ct * 16 + m];
      bO[ct][c].y = M2[(r + 1) * 48 + ct * 16 + m];
    }
    biasv[ct] = bo[ct * 16 + m];
  }

#if HAVE_ASYNC_LDS
  // double-buffered V tile: 16 rows x 12-dword pitch (16B aligned, no bank
  // conflicts on the b64 fragment reads)
  __shared__ float vsh[WPB][2 * 192];
  float* vb0 = &vsh[wave][0];
  float* vb1 = &vsh[wave][192];
  if (wgid < N_TILES)
    ASYNC_B128(Vws + (size_t)wgid * 128 + lane * 4,
               vb0 + (lane >> 1) * 12 + (lane & 1) * 4, 0);
  int parity = 0;
#endif

  for (int tile = wgid; tile < N_TILES; tile += TOTAL_WAVES) {
    const int nt = tile + TOTAL_WAVES;
    v2f a0, a1;
#if HAVE_ASYNC_LDS
    float* cur = parity ? vb1 : vb0;
    float* nxt = parity ? vb0 : vb1;
    if (nt < N_TILES) {
      ASYNC_B128(Vws + (size_t)nt * 128 + lane * 4,
                 nxt + (lane >> 1) * 12 + (lane & 1) * 4, 0);
      WAIT_ASYNC(1);
    } else {
      WAIT_ASYNC(0);
    }
    parity ^= 1;
    a0 = *(const v2f*)&cur[m * 12 + 0 + 2 * hi];
    a1 = *(const v2f*)&cur[m * 12 + 4 + 2 * hi];
#else
    const float* vb = Vws + (size_t)tile * 128;
    if (nt < N_TILES)
      __builtin_prefetch(Vws + (size_t)nt * 128 + lane * 4, 0, 3);
    a0 = *(const v2f*)&vb[m * 8 + 0 + 2 * hi];
    a1 = *(const v2f*)&vb[m * 8 + 4 + 2 * hi];
#endif

    v8f acc[3];
#pragma unroll
    for (int ct = 0; ct < 3; ++ct) {
#pragma unroll
      for (int i = 0; i < 8; ++i) acc[ct][i] = biasv[ct];
      acc[ct] = wmma_f32(a0, bO[ct][0], acc[ct]);
      acc[ct] = wmma_f32(a1, bO[ct][1], acc[ct]);
    }

    // streaming NT stores: out[(tile*16 + v + 8*hi)*48 + ct*16 + m]
    float* ob = out + (size_t)tile * 768;
#pragma unroll
    for (int ct = 0; ct < 3; ++ct)
#pragma unroll
      for (int v = 0; v < 8; ++v)
        __builtin_nontemporal_store(acc[ct][v],
                                    &ob[(v + 8 * hi) * 48 + ct * 16 + m]);
  }
}

// ---------------------------------------------------------------------------
extern "C" void kernel_launch(void* const* d_in, const int* in_sizes, int n_in,
                              void* d_out, int out_size, void* d_ws, size_t ws_size,
                              hipStream_t stream) {
  const float* x  = (const float*)d_in[0];
  const float* Wk = (const float*)d_in[1];
  const float* bk = (const float*)d_in[2];
  const float* Wq = (const float*)d_in[3];
  const float* bq = (const float*)d_in[4];
  const float* Wv = (const float*)d_in[5];
  const float* bv = (const float*)d_in[6];
  const float* Wo = (const float*)d_in[7];
  const float* bo = (const float*)d_in[8];

  float* Vws      = (float*)d_ws;                    // N*8 floats (32 MB)
  float* partials = Vws + (size_t)N_ROWS * 8;        // BLOCKS_MAIN*64 floats
  float* M2       = partials + BLOCKS_MAIN * 64;     // 8*48 floats

  pass1<<<BLOCKS_MAIN, 256, 0, stream>>>(x, Wk, bk, Wq, bq, Wv, bv, Vws, partials);
  pass_mid<<<1, 512, 0, stream>>>(partials, Wo, M2);
  pass2<<<BLOCKS_MAIN, 256, 0, stream>>>(Vws, M2, bo, (float*)d_out);
}